// TopologicalRegularizerAblation_23819888623784
// MI455X (gfx1250) — compile-verified
//
#include <hip/hip_runtime.h>
#include <hip/hip_bf16.h>

// PH0 (V-filtration, 4-connectivity) top-k lifetime loss for B=4 images of
// 256x256, matching the JAX reference:
//   f = 1 - prob*roi ; Kruskal over grid edges (w = max endpoint), elder rule;
//   top-5 lifetimes -> mean((l-0.5)^2) ; batch mean * 100.
//
// Design for MI455X:
//  * one 1024-thread (32-wave) workgroup per image
//  * stable LSD radix sort (4 x 8-bit passes) over 130,560 (key,edge) pairs;
//    stability via per-chunk digit prefix tables; in-chunk ranks computed from
//    an LDS-staged tile (async-to-LDS on gfx1250, tracked by ASYNCcnt)
//  * all scratch (~14 MB) is L2-resident (192 MB L2) -> sort passes never hit HBM
//  * single-lane union-find sweep with path halving + running top-5
//  * deterministic fixed-order final reduction kernel

namespace {
constexpr int   kB         = 4;
constexpr int   kH         = 256;
constexpr int   kW         = 256;
constexpr int   kN         = kH * kW;              // 65536
constexpr int   kER        = kH * (kW - 1);        // 65280 right edges
constexpr int   kE         = kER + (kH - 1) * kW;  // 130560 total edges
constexpr int   kChunkLog  = 7;
constexpr int   kChunk     = 1 << kChunkLog;       // 128
constexpr int   kNChunk    = kE / kChunk;          // 1020 (exact)
constexpr int   kTileChunks= 60;
constexpr int   kTile      = kTileChunks * kChunk; // 7680 keys (30 KB LDS)
constexpr int   kNTiles    = kNChunk / kTileChunks;// 17 (exact)
constexpr int   kThreads   = 1024;
constexpr int   kTopK      = 5;
constexpr float kTargetLife= 0.5f;
constexpr float kLossScale = 100.0f;
} // namespace

#if defined(__AMDGCN__) && __has_builtin(__builtin_amdgcn_global_load_async_to_lds_b32)
#define PH_ASYNC 1
#else
#define PH_ASYNC 0
#endif

#if PH_ASYNC
typedef __attribute__((address_space(1))) int ph_gint_t;   // global
typedef __attribute__((address_space(3))) int ph_lint_t;   // LDS
#endif

__device__ __forceinline__ void ph_async_copy_b32(const unsigned* g, unsigned* l) {
#if PH_ASYNC
  // per-lane async copy global -> LDS, tracked by ASYNCcnt
  __builtin_amdgcn_global_load_async_to_lds_b32(
      (ph_gint_t*)g, (ph_lint_t*)l, /*imm offset*/ 0, /*cpol*/ 0);
#else
  *l = *g;
#endif
}

__device__ __forceinline__ void ph_async_wait0() {
#if PH_ASYNC
#if __has_builtin(__builtin_amdgcn_s_wait_asynccnt)
  __builtin_amdgcn_s_wait_asynccnt(0);
#else
  asm volatile("s_wait_asynccnt 0" ::: "memory");
#endif
#endif
}

__device__ __forceinline__ void ph_edge_uv(int e, int& u, int& v) {
  if (e < kER) {               // right edge: idx2d[:, :-1] row-major
    int r = e / (kW - 1);
    int c = e - r * (kW - 1);
    u = r * kW + c;
    v = u + 1;
  } else {                     // down edge: idx2d[:-1, :] row-major == linear
    u = e - kER;
    v = u + kW;
  }
}

__global__ __launch_bounds__(kThreads)
void ph0_image_kernel(const float* __restrict__ prob,
                      const float* __restrict__ roi,
                      float* __restrict__ fAll,
                      unsigned* __restrict__ parentAll,
                      unsigned* __restrict__ keyA, unsigned* __restrict__ valA,
                      unsigned* __restrict__ keyB, unsigned* __restrict__ valB,
                      unsigned* __restrict__ chistAll,
                      float* __restrict__ losses)
{
  const int b   = blockIdx.x;
  const int tid = threadIdx.x;

  float*    f      = fAll      + (size_t)b * kN;
  unsigned* parent = parentAll + (size_t)b * kN;
  unsigned* kAp    = keyA + (size_t)b * kE;
  unsigned* vAp    = valA + (size_t)b * kE;
  unsigned* kBp    = keyB + (size_t)b * kE;
  unsigned* vBp    = valB + (size_t)b * kE;
  unsigned* ch     = chistAll + (size_t)b * 256 * kNChunk;

  __shared__ unsigned s_keys[kTile];   // 30 KB
  __shared__ unsigned s_base[256];     // bucket bases (exclusive over digits)
  __shared__ unsigned s_tot[256];      // per-digit totals

  // ---- phase 0: filtration values + union-find parents -------------------
  for (int i = tid; i < kN; i += kThreads) {
    float p = prob[(size_t)b * kN + i];
    float m = roi [(size_t)b * kN + i];
    f[i] = 1.0f - p * m;
    parent[i] = (unsigned)i;
  }
  __syncthreads();

  // ---- phase 1: edge keys (weight bits) + edge ids -----------------------
  for (int e = tid; e < kE; e += kThreads) {
    int u, v;
    ph_edge_uv(e, u, v);
    float w = fmaxf(f[u], f[v]);           // all values >= 0 -> bits are order-preserving
    kAp[e] = __float_as_uint(w);
    vAp[e] = (unsigned)e;
  }
  __syncthreads();

  // ---- phase 2: stable LSD radix sort, 4 passes of 8 bits ----------------
  unsigned* sk = kAp; unsigned* sv = vAp;
  unsigned* dk = kBp; unsigned* dv = vBp;

  for (int pass = 0; pass < 4; ++pass) {
    const int shift = pass * 8;

    // 2a. zero per-chunk digit histogram [digit][chunk]
    for (int i = tid; i < 256 * kNChunk; i += kThreads) ch[i] = 0u;
    __syncthreads();

    // 2b. histogram (order-independent integer atomics -> deterministic)
    for (int e = tid; e < kE; e += kThreads) {
      unsigned d = (sk[e] >> shift) & 0xFFu;
      atomicAdd(&ch[d * kNChunk + (e >> kChunkLog)], 1u);
    }
    __syncthreads();

    // 2c. exclusive scan over chunks, per digit (one thread per digit)
    if (tid < 256) {
      unsigned run = 0;
      unsigned* row = ch + (size_t)tid * kNChunk;
      for (int c = 0; c < kNChunk; ++c) { unsigned t = row[c]; row[c] = run; run += t; }
      s_tot[tid] = run;
    }
    __syncthreads();

    // 2d. exclusive scan over the 256 digit totals
    if (tid == 0) {
      unsigned run = 0;
      for (int d = 0; d < 256; ++d) { unsigned t = s_tot[d]; s_base[d] = run; run += t; }
    }
    __syncthreads();

    // 2e. stable scatter, tile by tile; keys staged into LDS via async copy
    for (int t = 0; t < kNTiles; ++t) {
      const int tb = t * kTile;
      for (int i = tid; i < kTile; i += kThreads) {
        ph_async_copy_b32(sk + tb + i, &s_keys[i]);   // gfx1250: global_load_async_to_lds_b32
        __builtin_prefetch(sv + tb + i, 0, 1);        // gfx1250: global_prefetch_b8 (payloads)
      }
      ph_async_wait0();                               // s_wait_asynccnt 0
      __syncthreads();

      for (int i = tid; i < kTile; i += kThreads) {
        const unsigned key = s_keys[i];
        const unsigned d   = (key >> shift) & 0xFFu;
        const int cbase    = i & ~(kChunk - 1);
        unsigned r = 0;                                // stable in-chunk rank
        for (int j = cbase; j < i; ++j)
          r += (((s_keys[j] >> shift) & 0xFFu) == d) ? 1u : 0u;
        const int e = tb + i;
        const unsigned pos = s_base[d] + ch[d * kNChunk + (e >> kChunkLog)] + r;
        dk[pos] = key;
        dv[pos] = sv[e];
      }
      __syncthreads();
    }

    unsigned* tk = sk; sk = dk; dk = tk;   // ping-pong (4 passes -> ends in A)
    unsigned* tv = sv; sv = dv; dv = tv;
  }
  __syncthreads();

  // ---- phase 3: sequential Kruskal sweep + running top-5 (lane 0) --------
  if (tid == 0) {
    float top[kTopK];
#pragma unroll
    for (int k = 0; k < kTopK; ++k) top[k] = -1.0f;   // lifetimes are >= 0

    for (int e = 0; e < kE; ++e) {
      const unsigned key = sk[e];
      const int id = (int)sv[e];
      int u, v;
      ph_edge_uv(id, u, v);

      // find with path halving
      int ru = u;
      for (;;) { unsigned p = parent[ru]; if ((int)p == ru) break;
                 unsigned gp = parent[p]; parent[ru] = gp; ru = (int)gp; }
      int rv = v;
      for (;;) { unsigned p = parent[rv]; if ((int)p == rv) break;
                 unsigned gp = parent[p]; parent[rv] = gp; rv = (int)gp; }

      if (ru != rv) {
        const float fu = f[ru];
        const float fv = f[rv];
        const bool older_u = (fu < fv) || ((fu == fv) && (ru < rv));  // elder rule
        const int keep = older_u ? ru : rv;
        const int die  = older_u ? rv : ru;
        const float w    = __uint_as_float(key);     // death value = edge weight
        const float life = w - f[die];               // death - birth
        parent[die] = (unsigned)keep;
        parent[u]   = (unsigned)keep;
        parent[v]   = (unsigned)keep;

        // keep the 5 largest lifetimes
        int mi = 0; float mv = top[0];
#pragma unroll
        for (int k = 1; k < kTopK; ++k) if (top[k] < mv) { mv = top[k]; mi = k; }
        if (life > mv) top[mi] = life;
      }
    }

    // 65535 merges >= 5 -> n_use == 5 always
    float acc = 0.0f;
#pragma unroll
    for (int k = 0; k < kTopK; ++k) {
      float d = top[k] - kTargetLife;
      acc += d * d;
    }
    losses[b] = acc / (float)kTopK;
  }
}

__global__ void ph0_finalize_kernel(const float* __restrict__ losses,
                                    float* __restrict__ out)
{
  // fixed-order sum -> bitwise deterministic
  float s = 0.0f;
  for (int b = 0; b < kB; ++b) s += losses[b];
  out[0] = (s / (float)kB) * kLossScale;
}

extern "C" void kernel_launch(void* const* d_in, const int* in_sizes, int n_in,
                              void* d_out, int out_size, void* d_ws, size_t ws_size,
                              hipStream_t stream)
{
  (void)in_sizes; (void)n_in; (void)out_size; (void)ws_size;

  const float* prob = (const float*)d_in[0];   // (B,1,H,W) float32
  const float* roi  = (const float*)d_in[1];   // (B,1,H,W) float32
  float* out = (float*)d_out;                  // scalar float32

  // workspace carve-up (all 256B-aligned); total ~14.3 MB, L2-resident
  char* ws = (char*)d_ws;
  size_t off = 0;
  auto carve = [&](size_t bytes) -> void* {
    void* p = ws + off;
    off += (bytes + 255u) & ~(size_t)255u;
    return p;
  };
  float*    f      = (float*)   carve((size_t)kB * kN * sizeof(float));
  unsigned* parent = (unsigned*)carve((size_t)kB * kN * sizeof(unsigned));
  unsigned* keyA   = (unsigned*)carve((size_t)kB * kE * sizeof(unsigned));
  unsigned* valA   = (unsigned*)carve((size_t)kB * kE * sizeof(unsigned));
  unsigned* keyB   = (unsigned*)carve((size_t)kB * kE * sizeof(unsigned));
  unsigned* valB   = (unsigned*)carve((size_t)kB * kE * sizeof(unsigned));
  unsigned* chist  = (unsigned*)carve((size_t)kB * 256 * kNChunk * sizeof(unsigned));
  float*    losses = (float*)   carve((size_t)kB * sizeof(float));

  ph0_image_kernel<<<dim3(kB), dim3(kThreads), 0, stream>>>(
      prob, roi, f, parent, keyA, valA, keyB, valB, chist, losses);
  ph0_finalize_kernel<<<dim3(1), dim3(1), 0, stream>>>(losses, out);
}